// PlutchikEmotionModelV2_82394652607015
// MI455X (gfx1250) — compile-verified
//
#include <hip/hip_runtime.h>
#include <hip/hip_bf16.h>
#include <math.h>

typedef __attribute__((ext_vector_type(2))) float v2f;
typedef __attribute__((ext_vector_type(8))) float v8f;

namespace {
constexpr int E = 8, H = 768, NH = 4, OUT = 128, HH = 32, AUX = 128;
constexpr int B = 64, S = 512, HD = H / NH;            // HD = 192
constexpr int EN = E * NH;                             // 32
// workspace layout (float offsets)
constexpr size_t WS_QK  = 0;                           // EN*H      = 24576
constexpr size_t WS_ATT = WS_QK + (size_t)EN * H;      // E*B*NH*S  = 1048576
constexpr size_t WS_CTX = WS_ATT + (size_t)E*B*NH*S;   // B*EN*H    = 1572864
constexpr size_t WS_PO  = WS_CTX + (size_t)B*EN*H;     // E*B*H     = 393216
constexpr size_t WS_PR  = WS_PO  + (size_t)E*B*H;      // E*B*OUT   = 65536
}

__device__ __forceinline__ v8f wmma4(v2f a, v2f b, v8f c) {
  // D = A(16x4 f32) * B(4x16 f32) + C(16x16 f32)
  return __builtin_amdgcn_wmma_f32_16x16x4_f32(false, a, false, b, (short)0, c,
                                               false, false);
}
__device__ __forceinline__ float gelu_exact(float x) {
  return 0.5f * x * (1.0f + erff(x * 0.70710678118654752f));
}

// ---- Stage 1: qk[en][h] = scale * sum_d Wk[e][h][n*HD+d] * q[e][n*HD+d] ----
__global__ void k_qk(const float* __restrict__ Wk, const float* __restrict__ q,
                     float* __restrict__ qk) {
  const int en = blockIdx.x, e = en >> 2, n = en & 3;
  __shared__ float qs[HD];
  for (int d = threadIdx.x; d < HD; d += blockDim.x) qs[d] = q[e * H + n * HD + d];
  __syncthreads();
  const float scale = rsqrtf((float)HD);
  for (int h = threadIdx.x; h < H; h += blockDim.x) {
    const float* wrow = Wk + ((size_t)e * H + h) * H + n * HD;
    float acc = 0.f;
    for (int d = 0; d < HD; ++d) acc += wrow[d] * qs[d];
    qk[(size_t)en * H + h] = acc * scale;
  }
}

// ---- Stage 2: logits (32768 x 32) = HS @ qk^T, masked -> att buffer -------
// One wave per 16-token M tile; each wave computes BOTH en N-tiles so the
// dominant HS (A) fragment is loaded once per k-step.
__global__ __launch_bounds__(64) void k_logits(const float* __restrict__ hs,
                                               const float* __restrict__ qk,
                                               const int* __restrict__ amask,
                                               float* __restrict__ att) {
  const int lane = threadIdx.x & 31, wave = threadIdx.x >> 5;
  const int hi = lane >> 4, lo = lane & 15;
  const int row0 = blockIdx.x * 32 + wave * 16;        // token tile base
  const float* arow  = hs + (size_t)(row0 + lo) * H;   // A: token row
  const float* brow0 = qk + (size_t)lo * H;            // B: en 0..15
  const float* brow1 = qk + (size_t)(16 + lo) * H;     // B: en 16..31
  v8f c0 = {}, c1 = {};
#pragma unroll 4
  for (int k0 = 0; k0 < H; k0 += 4) {
    v2f a, b0, b1;
    a.x  = arow[k0 + 2 * hi];
    a.y  = arow[k0 + 2 * hi + 1];
    b0.x = brow0[k0 + hi];
    b0.y = brow0[k0 + hi + 2];
    b1.x = brow1[k0 + hi];
    b1.y = brow1[k0 + hi + 2];
    c0 = wmma4(a, b0, c0);
    c1 = wmma4(a, b1, c1);
  }
#pragma unroll
  for (int half = 0; half < 2; ++half) {
    const int ncol = half * 16 + lo;
    const int e = ncol >> 2, n = ncol & 3;
    const v8f& c = half ? c1 : c0;
#pragma unroll
    for (int r = 0; r < 8; ++r) {
      const int token = row0 + r + 8 * hi;
      const int bb = token >> 9, ss = token & (S - 1);
      float v = c[r];
      if (amask[bb * S + ss] == 0) v = -INFINITY;
      att[(((size_t)e * B + bb) * NH + n) * S + ss] = v;
    }
  }
}

// ---- Stage 3: softmax over S (row length 512) -----------------------------
__global__ __launch_bounds__(256) void k_softmax(float* __restrict__ att) {
  float* row = att + (size_t)blockIdx.x * S;
  const int t = threadIdx.x;
  float a = row[t], b = row[t + 256];
  __shared__ float red[256];
  red[t] = fmaxf(a, b); __syncthreads();
  for (int o = 128; o > 0; o >>= 1) { if (t < o) red[t] = fmaxf(red[t], red[t + o]); __syncthreads(); }
  const float mx = red[0]; __syncthreads();
  const float ea = __expf(a - mx), eb = __expf(b - mx);
  red[t] = ea + eb; __syncthreads();
  for (int o = 128; o > 0; o >>= 1) { if (t < o) red[t] += red[t + o]; __syncthreads(); }
  const float inv = 1.0f / red[0];
  row[t] = ea * inv; row[t + 256] = eb * inv;
}

// ---- Stage 4: ctx[b][en][h] = att_rows(32x512) @ HS_b(512x768) ------------
// One wave per 16-wide h tile; each wave computes BOTH en M-tiles so the
// strided HS (B) fragment is loaded once per k-step.
__global__ __launch_bounds__(64) void k_ctx(const float* __restrict__ att,
                                            const float* __restrict__ hs,
                                            float* __restrict__ ctx) {
  const int lane = threadIdx.x & 31, wave = threadIdx.x >> 5;
  const int hi = lane >> 4, lo = lane & 15;
  const int bb = blockIdx.y;
  const int hcol = blockIdx.x * 32 + wave * 16 + lo;
  const int m0 = lo, m1 = 16 + lo;                     // en rows for A loads
  const float* arow0 = att + (((size_t)(m0 >> 2) * B + bb) * NH + (m0 & 3)) * S;
  const float* arow1 = att + (((size_t)(m1 >> 2) * B + bb) * NH + (m1 & 3)) * S;
  const float* hsb = hs + (size_t)bb * S * H;
  v8f c0 = {}, c1 = {};
#pragma unroll 4
  for (int k0 = 0; k0 < S; k0 += 4) {
    v2f a0, a1, b;
    b.x  = hsb[(size_t)(k0 + hi) * H + hcol];
    b.y  = hsb[(size_t)(k0 + hi + 2) * H + hcol];
    a0.x = arow0[k0 + 2 * hi];
    a0.y = arow0[k0 + 2 * hi + 1];
    a1.x = arow1[k0 + 2 * hi];
    a1.y = arow1[k0 + 2 * hi + 1];
    c0 = wmma4(a0, b, c0);
    c1 = wmma4(a1, b, c1);
  }
#pragma unroll
  for (int r = 0; r < 8; ++r) {
    const int en0 = r + 8 * hi;
    ctx[((size_t)bb * EN + en0) * H + hcol] = c0[r];
    ctx[((size_t)bb * EN + en0 + 16) * H + hcol] = c1[r];
  }
}

// ---- Stage 5a: pooled[e][b][n*HD+d] = ctx rows(b) @ Wv[e][:, head n] ------
// Each wave owns one 16-row b tile and computes two d N-tiles (A reused).
__global__ __launch_bounds__(128) void k_pooled(const float* __restrict__ ctx,
                                                const float* __restrict__ Wv,
                                                float* __restrict__ pooled) {
  const int lane = threadIdx.x & 31, wave = threadIdx.x >> 5; // wave = b tile
  const int hi = lane >> 4, lo = lane & 15;
  const int en = blockIdx.y, e = en >> 2, n = en & 3;
  const int dcol0 = blockIdx.x * 32 + lo, dcol1 = dcol0 + 16;
  const float* arow = ctx + ((size_t)(wave * 16 + lo) * EN + en) * H;
  const float* wv0 = Wv + (size_t)e * H * H + n * HD + dcol0;
  const float* wv1 = Wv + (size_t)e * H * H + n * HD + dcol1;
  v8f c0 = {}, c1 = {};
#pragma unroll 4
  for (int k0 = 0; k0 < H; k0 += 4) {
    v2f a, b0, b1;
    a.x  = arow[k0 + 2 * hi];
    a.y  = arow[k0 + 2 * hi + 1];
    b0.x = wv0[(size_t)(k0 + hi) * H];
    b0.y = wv0[(size_t)(k0 + hi + 2) * H];
    b1.x = wv1[(size_t)(k0 + hi) * H];
    b1.y = wv1[(size_t)(k0 + hi + 2) * H];
    c0 = wmma4(a, b0, c0);
    c1 = wmma4(a, b1, c1);
  }
#pragma unroll
  for (int r = 0; r < 8; ++r) {
    const int bbr = wave * 16 + r + 8 * hi;
    pooled[((size_t)e * B + bbr) * H + n * HD + dcol0] = c0[r];
    pooled[((size_t)e * B + bbr) * H + n * HD + dcol1] = c1[r];
  }
}

// ---- Stage 5b: proj[e][b][o] = pooled(64x768) @ Wo[e](768x128) + bo -------
// Each wave owns one 16-row b tile and computes two o N-tiles (A reused).
__global__ __launch_bounds__(128) void k_proj(const float* __restrict__ pooled,
                                              const float* __restrict__ Wo,
                                              const float* __restrict__ bo,
                                              float* __restrict__ proj) {
  const int lane = threadIdx.x & 31, wave = threadIdx.x >> 5; // wave = b tile
  const int hi = lane >> 4, lo = lane & 15;
  const int e = blockIdx.y;
  const int ocol0 = blockIdx.x * 32 + lo, ocol1 = ocol0 + 16;
  const float* arow = pooled + ((size_t)e * B + wave * 16 + lo) * H;
  const float* wo0 = Wo + (size_t)e * H * OUT + ocol0;
  const float* wo1 = Wo + (size_t)e * H * OUT + ocol1;
  v8f c0 = {}, c1 = {};
#pragma unroll 4
  for (int k0 = 0; k0 < H; k0 += 4) {
    v2f a, b0, b1;
    a.x  = arow[k0 + 2 * hi];
    a.y  = arow[k0 + 2 * hi + 1];
    b0.x = wo0[(size_t)(k0 + hi) * OUT];
    b0.y = wo0[(size_t)(k0 + hi + 2) * OUT];
    b1.x = wo1[(size_t)(k0 + hi) * OUT];
    b1.y = wo1[(size_t)(k0 + hi + 2) * OUT];
    c0 = wmma4(a, b0, c0);
    c1 = wmma4(a, b1, c1);
  }
  const float bias0 = bo[e * OUT + ocol0];
  const float bias1 = bo[e * OUT + ocol1];
#pragma unroll
  for (int r = 0; r < 8; ++r) {
    const int bbr = wave * 16 + r + 8 * hi;
    proj[((size_t)e * B + bbr) * OUT + ocol0] = c0[r] + bias0;
    proj[((size_t)e * B + bbr) * OUT + ocol1] = c1[r] + bias1;
  }
}

// ---- Stage 6: layernorm + MLP head -> scores[b][e] ------------------------
__global__ __launch_bounds__(128) void k_head(const float* __restrict__ proj,
    const float* __restrict__ ln_g, const float* __restrict__ ln_b,
    const float* __restrict__ hW1, const float* __restrict__ hb1,
    const float* __restrict__ hW2, const float* __restrict__ hb2,
    const float* __restrict__ temp, float* __restrict__ out) {
  const int e = blockIdx.x >> 6, b = blockIdx.x & 63;
  const int t = threadIdx.x;                      // 0..127 == OUT
  __shared__ float red[OUT];
  __shared__ float sfe[OUT];
  __shared__ float shm[HH];
  const float x = proj[((size_t)e * B + b) * OUT + t];
  red[t] = x; __syncthreads();
  for (int o = 64; o > 0; o >>= 1) { if (t < o) red[t] += red[t + o]; __syncthreads(); }
  const float mu = red[0] * (1.0f / OUT); __syncthreads();
  const float d = x - mu;
  red[t] = d * d; __syncthreads();
  for (int o = 64; o > 0; o >>= 1) { if (t < o) red[t] += red[t + o]; __syncthreads(); }
  const float var = red[0] * (1.0f / OUT);
  sfe[t] = d * rsqrtf(var + 1e-5f) * ln_g[e * OUT + t] + ln_b[e * OUT + t];
  __syncthreads();
  if (t < HH) {
    float acc = hb1[e * HH + t];
    for (int o = 0; o < OUT; ++o) acc += sfe[o] * hW1[((size_t)e * OUT + o) * HH + t];
    shm[t] = gelu_exact(acc);
  }
  __syncthreads();
  if (t == 0) {
    float elog = hb2[e];
    for (int c = 0; c < HH; ++c) elog += shm[c] * hW2[e * HH + c];
    const float tc = fminf(fmaxf(temp[e], 0.5f), 5.0f);
    out[b * E + e] = 1.0f / (1.0f + expf(-elog * tc));
  }
}

// ---- Stage 7: cls token -> conf + cls_logits ------------------------------
__global__ __launch_bounds__(128) void k_cls(const float* __restrict__ hs,
    const float* __restrict__ cW1, const float* __restrict__ cb1,
    const float* __restrict__ cW2, const float* __restrict__ cb2,
    const float* __restrict__ clsW1, const float* __restrict__ clsb1,
    const float* __restrict__ clsW2, const float* __restrict__ clsb2,
    float* __restrict__ out) {
  const int b = blockIdx.x, t = threadIdx.x;      // t = 0..127 == AUX
  __shared__ float scls[H];
  __shared__ float s1[AUX];
  __shared__ float s2[AUX];
  for (int h = t; h < H; h += 128) scls[h] = hs[(size_t)b * S * H + h];
  __syncthreads();
  float a1 = cb1[t], a2 = clsb1[t];
  for (int h = 0; h < H; ++h) {
    const float cv = scls[h];
    a1 += cv * cW1[h * AUX + t];
    a2 += cv * clsW1[h * AUX + t];
  }
  s1[t] = gelu_exact(a1) * cW2[t];
  s2[t] = gelu_exact(a2);
  __syncthreads();
  for (int o = 64; o > 0; o >>= 1) { if (t < o) s1[t] += s1[t + o]; __syncthreads(); }
  if (t == 0) {
    const float cf = s1[0] + cb2[0];
    out[B * E + b] = 1.0f / (1.0f + expf(-cf));    // conf at offset 512
  }
  if (t < E) {
    float cl = clsb2[t];
    for (int j = 0; j < AUX; ++j) cl += s2[j] * clsW2[j * E + t];
    out[B * E + B + b * E + t] = cl;               // cls_logits at offset 576
  }
}

extern "C" void kernel_launch(void* const* d_in, const int* in_sizes, int n_in,
                              void* d_out, int out_size, void* d_ws, size_t ws_size,
                              hipStream_t stream) {
  const float* HS    = (const float*)d_in[0];
  const int*   amask = (const int*)  d_in[1];
  const float* q     = (const float*)d_in[2];
  const float* Wk    = (const float*)d_in[3];
  const float* Wv    = (const float*)d_in[4];
  const float* Wo    = (const float*)d_in[5];
  const float* bo    = (const float*)d_in[6];
  const float* ln_g  = (const float*)d_in[7];
  const float* ln_b  = (const float*)d_in[8];
  const float* hW1   = (const float*)d_in[9];
  const float* hb1   = (const float*)d_in[10];
  const float* hW2   = (const float*)d_in[11];
  const float* hb2   = (const float*)d_in[12];
  const float* temp  = (const float*)d_in[13];
  const float* cW1   = (const float*)d_in[14];
  const float* cb1   = (const float*)d_in[15];
  const float* cW2   = (const float*)d_in[16];
  const float* cb2   = (const float*)d_in[17];
  const float* clsW1 = (const float*)d_in[18];
  const float* clsb1 = (const float*)d_in[19];
  const float* clsW2 = (const float*)d_in[20];
  const float* clsb2 = (const float*)d_in[21];
  float* out = (float*)d_out;
  float* ws  = (float*)d_ws;

  float* qk     = ws + WS_QK;
  float* att    = ws + WS_ATT;
  float* ctx    = ws + WS_CTX;
  float* pooled = ws + WS_PO;
  float* proj   = ws + WS_PR;

  k_qk     <<<EN, 256, 0, stream>>>(Wk, q, qk);
  k_logits <<<(B * S) / 32, 64, 0, stream>>>(HS, qk, amask, att);
  k_softmax<<<E * B * NH, 256, 0, stream>>>(att);
  k_ctx    <<<dim3(H / 32, B), 64, 0, stream>>>(att, HS, ctx);
  k_pooled <<<dim3(HD / 32, EN), 128, 0, stream>>>(ctx, Wv, pooled);
  k_proj   <<<dim3(OUT / 32, E), 128, 0, stream>>>(pooled, Wo, bo, proj);
  k_head   <<<E * B, 128, 0, stream>>>(proj, ln_g, ln_b, hW1, hb1, hW2, hb2, temp, out);
  k_cls    <<<B, 128, 0, stream>>>(HS, cW1, cb1, cW2, cb2, clsW1, clsb1, clsW2, clsb2, out);
}